// Encoder_326417514604
// MI455X (gfx1250) — compile-verified
//
#include <hip/hip_runtime.h>
#include <math.h>

#define N_NODES    100000
#define N_EDGES    1600000
#define H_DIM      64
#define N_LAYERS   3
#define NUM_GRAPHS 256

typedef __attribute__((ext_vector_type(2))) float v2f;
typedef __attribute__((ext_vector_type(8))) float v8f;

// D = A(16x4 f32) * B(4x16 f32) + C(16x16 f32), full-precision f32 WMMA.
__device__ __forceinline__ v8f wmma4(v2f a, v2f b, v8f c) {
  return __builtin_amdgcn_wmma_f32_16x16x4_f32(false, a, false, b, (short)0, c,
                                               false, false);
}

// ---------------------------------------------------------------------------
// Pack a [64 x (nGroups*16)] B matrix into WMMA-ready per-lane fragments:
//   frag f = (kk*nGroups + g)*32 + lane  ->  v2f{ B[4kk+2lh][g*16+ln],
//                                                B[4kk+2lh+1][g*16+ln] }
// transposed=1 means the logical B is Wsrc^T (Wsrc row-major [cols][64]).
// ---------------------------------------------------------------------------
__global__ void pack_b(const float* __restrict__ Wsrc, float* __restrict__ pack,
                       int nGroups, int transposed) {
  int tid = blockIdx.x * blockDim.x + threadIdx.x;
  int total = 16 * nGroups * 32;
  if (tid >= total) return;
  int lane = tid & 31;
  int t = tid >> 5;
  int g = t % nGroups, kk = t / nGroups;
  int ln = lane & 15, lh = lane >> 4;
  int c = g * 16 + ln;
  int k0 = 4 * kk + 2 * lh;
  v2f b;
  if (transposed) {           // B[k][c] = Wsrc[c][k]
    b.x = Wsrc[c * 64 + k0];
    b.y = Wsrc[c * 64 + k0 + 1];
  } else {                    // B[k][c] = Wsrc[k][c]
    b.x = Wsrc[k0 * 64 + c];
    b.y = Wsrc[(k0 + 1) * 64 + c];
  }
  ((v2f*)pack)[tid] = b;
}

// ---------------------------------------------------------------------------
// m[N,64] = h[N,64] @ W[64,64], B from packed fragments (1 coalesced b64 per
// fragment, shared across all waves via L1). One wave per 16-row tile.
// ---------------------------------------------------------------------------
__global__ void gemm_m(const float* __restrict__ h,
                       const v2f* __restrict__ packW, float* __restrict__ m,
                       int nTiles) {
  int wave = (int)((blockIdx.x * blockDim.x + threadIdx.x) >> 5);
  int lane = threadIdx.x & 31;
  if (wave >= nTiles) return;                 // wave-uniform: EXEC all-1s
  const int r0 = wave * 16;
  const int ln = lane & 15, lh = lane >> 4;

  v2f a[16];
  const v2f* hp = (const v2f*)(h + (size_t)(r0 + ln) * H_DIM);
#pragma unroll
  for (int kk = 0; kk < 16; ++kk) a[kk] = hp[2 * kk + lh];

  v8f acc[4];
#pragma unroll
  for (int j = 0; j < 4; ++j) {
    v8f z = {};
    acc[j] = z;
  }

#pragma unroll
  for (int kk = 0; kk < 16; ++kk) {
#pragma unroll
    for (int j = 0; j < 4; ++j) {
      v2f b = packW[(kk * 4 + j) * 32 + lane];
      acc[j] = wmma4(a[kk], b, acc[j]);
    }
  }

#pragma unroll
  for (int j = 0; j < 4; ++j)
#pragma unroll
    for (int v = 0; v < 8; ++v)
      m[(size_t)(r0 + v + 8 * lh) * H_DIM + j * 16 + ln] = acc[j][v];
}

// ---------------------------------------------------------------------------
// agg[dst] += m[src] over all edges. 16 lanes per edge, b128 gather,
// hardware f32 atomics (L2-resident, relaxed/agent scope).
// ---------------------------------------------------------------------------
__global__ void scatter_add(const float* __restrict__ m,
                            const int* __restrict__ ei,
                            float* __restrict__ agg, int nEdges) {
  int tid = blockIdx.x * blockDim.x + threadIdx.x;
  int e = tid >> 4;
  if (e >= nEdges) return;
  int q = tid & 15;
  int src = ei[e];
  int dst = ei[nEdges + e];
  float4 val = ((const float4*)(m + (size_t)src * H_DIM))[q];
  float* ap = agg + (size_t)dst * H_DIM + q * 4;
  __hip_atomic_fetch_add(ap + 0, val.x, __ATOMIC_RELAXED, __HIP_MEMORY_SCOPE_AGENT);
  __hip_atomic_fetch_add(ap + 1, val.y, __ATOMIC_RELAXED, __HIP_MEMORY_SCOPE_AGENT);
  __hip_atomic_fetch_add(ap + 2, val.z, __ATOMIC_RELAXED, __HIP_MEMORY_SCOPE_AGENT);
  __hip_atomic_fetch_add(ap + 3, val.w, __ATOMIC_RELAXED, __HIP_MEMORY_SCOPE_AGENT);
}

// ---------------------------------------------------------------------------
// Fused GRU cell, in-place on h (row-local -> safe). Packed W_ih^T / W_hh^T
// fragment tables (48 KB each) are staged into LDS once per block; the inner
// loop feeds 6 WMMA streams from conflict-free ds_load_b64 (256B per wave per
// fragment). 384 v_wmma per wave; gate math in the C-matrix register layout.
// ---------------------------------------------------------------------------
#define NFRAG_G (16 * 12 * 32)   // 6144 v2f fragments per gate matrix (48 KB)

__global__ void gru_fused(float* __restrict__ h, const float* __restrict__ agg,
                          const v2f* __restrict__ pIh,
                          const v2f* __restrict__ pHh,
                          const float* __restrict__ bih,
                          const float* __restrict__ bhh, int nTiles) {
  __shared__ v2f sIh[NFRAG_G];
  __shared__ v2f sHh[NFRAG_G];

  // cooperative stage: 96 KB global -> LDS, coalesced b64 + ds_store_b64
  for (int i = threadIdx.x; i < NFRAG_G; i += 256) {
    sIh[i] = pIh[i];
    sHh[i] = pHh[i];
  }
  __syncthreads();

  int wave = (int)((blockIdx.x * blockDim.x + threadIdx.x) >> 5);
  int lane = threadIdx.x & 31;
  if (wave >= nTiles) return;                 // wave-uniform
  const int r0 = wave * 16;
  const int ln = lane & 15, lh = lane >> 4;

  v2f aA[16], aH[16];
  const v2f* ap = (const v2f*)(agg + (size_t)(r0 + ln) * H_DIM);
  const v2f* hp = (const v2f*)(h + (size_t)(r0 + ln) * H_DIM);
#pragma unroll
  for (int kk = 0; kk < 16; ++kk) {
    aA[kk] = ap[2 * kk + lh];
    aH[kk] = hp[2 * kk + lh];
  }

#pragma unroll 1
  for (int j = 0; j < 4; ++j) {
    const int c = j * 16 + ln;  // column within one 64-wide gate group
    float bir = bih[c], biz = bih[64 + c], bin = bih[128 + c];
    float bhr = bhh[c], bhz = bhh[64 + c], bhn = bhh[128 + c];
    v8f gi_r, gi_z, gi_n, gh_r, gh_z, gh_n;
#pragma unroll
    for (int v = 0; v < 8; ++v) {
      gi_r[v] = bir; gi_z[v] = biz; gi_n[v] = bin;
      gh_r[v] = bhr; gh_z[v] = bhz; gh_n[v] = bhn;
    }

    // gate-group fragment indices within the 12 column-groups: j, j+4, j+8
#pragma unroll
    for (int kk = 0; kk < 16; ++kk) {
      int base = kk * 12 * 32 + lane;
      v2f br = sIh[base + (j + 0) * 32];
      v2f bz = sIh[base + (j + 4) * 32];
      v2f bn = sIh[base + (j + 8) * 32];
      v2f cr = sHh[base + (j + 0) * 32];
      v2f cz = sHh[base + (j + 4) * 32];
      v2f cn = sHh[base + (j + 8) * 32];
      gi_r = wmma4(aA[kk], br, gi_r);
      gi_z = wmma4(aA[kk], bz, gi_z);
      gi_n = wmma4(aA[kk], bn, gi_n);
      gh_r = wmma4(aH[kk], cr, gh_r);
      gh_z = wmma4(aH[kk], cz, gh_z);
      gh_n = wmma4(aH[kk], cn, gh_n);
    }

#pragma unroll
    for (int v = 0; v < 8; ++v) {
      size_t idx = (size_t)(r0 + v + 8 * lh) * H_DIM + c;
      float hv = h[idx];                       // read-before-write, same wave
      float r = 1.0f / (1.0f + __expf(-(gi_r[v] + gh_r[v])));
      float z = 1.0f / (1.0f + __expf(-(gi_z[v] + gh_z[v])));
      float n = tanhf(gi_n[v] + r * gh_n[v]);
      h[idx] = (1.0f - z) * n + z * hv;
    }
  }
}

// ---------------------------------------------------------------------------
__global__ void zero_f32(float* __restrict__ p, int n) {
  int tid = blockIdx.x * blockDim.x + threadIdx.x;
  if (tid < n) p[tid] = 0.0f;
}

__global__ void pool(const float* __restrict__ h, const int* __restrict__ batch,
                     float* __restrict__ out, int nNodes) {
  int tid = blockIdx.x * blockDim.x + threadIdx.x;
  int n = tid >> 6;
  if (n >= nNodes) return;
  int d = tid & 63;
  int g = batch[n];
  __hip_atomic_fetch_add(out + (size_t)g * H_DIM + d, h[(size_t)n * H_DIM + d],
                         __ATOMIC_RELAXED, __HIP_MEMORY_SCOPE_AGENT);
}

// ---------------------------------------------------------------------------
extern "C" void kernel_launch(void* const* d_in, const int* in_sizes, int n_in,
                              void* d_out, int out_size, void* d_ws,
                              size_t ws_size, hipStream_t stream) {
  const float* x      = (const float*)d_in[0];
  const int*   ei     = (const int*)d_in[1];
  const int*   batch  = (const int*)d_in[2];
  const float* weight = (const float*)d_in[3];
  const float* Wih    = (const float*)d_in[4];
  const float* Whh    = (const float*)d_in[5];
  const float* bih    = (const float*)d_in[6];
  const float* bhh    = (const float*)d_in[7];
  float* out = (float*)d_out;

  const size_t NB = (size_t)N_NODES * H_DIM * sizeof(float);  // 25.6 MB
  const int NFRAG_W = 16 * 4 * 32;                            // 2048 frags/layer
  char* ws = (char*)d_ws;
  float* h    = (float*)(ws);
  float* m    = (float*)(ws + NB);
  float* agg  = (float*)(ws + 2 * NB);
  v2f* packW  = (v2f*)(ws + 3 * NB);                 // 3 * 16 KB
  v2f* packIh = packW + (size_t)N_LAYERS * NFRAG_W;  // 48 KB
  v2f* packHh = packIh + NFRAG_G;                    // 48 KB

  hipMemcpyAsync(h, x, NB, hipMemcpyDeviceToDevice, stream);
  for (int layer = 0; layer < N_LAYERS; ++layer)
    pack_b<<<(NFRAG_W + 255) / 256, 256, 0, stream>>>(
        weight + (size_t)layer * 64 * 64, (float*)(packW + (size_t)layer * NFRAG_W),
        4, 0);
  pack_b<<<(NFRAG_G + 255) / 256, 256, 0, stream>>>(Wih, (float*)packIh, 12, 1);
  pack_b<<<(NFRAG_G + 255) / 256, 256, 0, stream>>>(Whh, (float*)packHh, 12, 1);

  const int nTiles = N_NODES / 16;                    // 6250 (exact)
  const int tileBlocks = (nTiles * 32 + 255) / 256;   // 8 waves per block
  const int nFeat = N_NODES * H_DIM;                  // 6.4M

  for (int layer = 0; layer < N_LAYERS; ++layer) {
    gemm_m<<<tileBlocks, 256, 0, stream>>>(h, packW + (size_t)layer * NFRAG_W,
                                           m, nTiles);
    zero_f32<<<(nFeat + 255) / 256, 256, 0, stream>>>(agg, nFeat);
    scatter_add<<<(N_EDGES * 16 + 255) / 256, 256, 0, stream>>>(m, ei, agg,
                                                                N_EDGES);
    gru_fused<<<tileBlocks, 256, 0, stream>>>(h, agg, packIh, packHh, bih, bhh,
                                              nTiles);
  }

  zero_f32<<<(NUM_GRAPHS * H_DIM + 255) / 256, 256, 0, stream>>>(
      out, NUM_GRAPHS * H_DIM);
  pool<<<(N_NODES * H_DIM + 255) / 256, 256, 0, stream>>>(h, batch, out,
                                                          N_NODES);
}